// DiceLoss_49830210568812
// MI455X (gfx1250) — compile-verified
//
#include <hip/hip_runtime.h>

// Dice loss for pred/target f32[4, 8, 2097152]; scalar f32 output.
// Bandwidth-bound (512 MiB stream @ 23.3 TB/s ~= 23 us).
// Dot products ride V_WMMA_F32_16X16X4_F32: for the accumulator DIAGONAL,
// lane l's contribution is dot(A-pair(l), B-pair(l)), so any per-lane element
// assignment works as long as A and B use the same indices. We therefore load
// 4 consecutive floats per lane (global_load_b128, wave covers 512B) and issue
// two WMMAs (.lo/.hi) into two independent accumulators per iteration.

typedef __attribute__((ext_vector_type(2))) float v2f;
typedef __attribute__((ext_vector_type(4))) float v4f;
typedef __attribute__((ext_vector_type(8))) float v8f;

#define WAVES_PER_BLOCK 8
#define BPS 256           // blocks per (b,c) slice
#define NSLICES 32        // N*C = 4*8
#define CHUNK 128         // elements per wave-iteration (two 16x16x4 WMMAs)

__global__ __launch_bounds__(256) void dice_stage1(
    const float* __restrict__ pred, const float* __restrict__ targ,
    float* __restrict__ pDot, float* __restrict__ pSum,
    int chunksPerSlice, int P)
{
    const int slice = blockIdx.y;
    const int lane  = threadIdx.x & 31;
    // Force wave id scalar so loop control is SALU (s_cmp/s_cbranch) and
    // addresses become SADDR + per-lane voffset; EXEC stays all-ones.
    const int wave  = __builtin_amdgcn_readfirstlane(threadIdx.x >> 5);
    const long long sliceBase = (long long)slice * (long long)P;
    const int voff = lane * 4;                 // 16B per lane, coalesced

    v8f accLo = {};      // diagonal collects dot of .lo pairs
    v8f accHi = {};      // diagonal collects dot of .hi pairs
    v2f s2 = {0.0f, 0.0f};   // packed running sum of pred+target

    const int stride = gridDim.x * WAVES_PER_BLOCK;       // scalar
    for (int chunk = blockIdx.x * WAVES_PER_BLOCK + wave;
         chunk < chunksPerSlice; chunk += stride) {
        const float* pp = pred + sliceBase + (long long)chunk * CHUNK;
        const float* tp = targ + sliceBase + (long long)chunk * CHUNK;
        v4f a = __builtin_nontemporal_load((const v4f*)(pp + voff));
        v4f b = __builtin_nontemporal_load((const v4f*)(tp + voff));
        v2f alo = {a.x, a.y}, ahi = {a.z, a.w};
        v2f blo = {b.x, b.y}, bhi = {b.z, b.w};
        s2 = s2 + alo + ahi + blo + bhi;       // v_pk_add_f32 chain
        // D = A*B + C  (neg_a, A, neg_b, B, c_mod, C, reuse_a, reuse_b)
        accLo = __builtin_amdgcn_wmma_f32_16x16x4_f32(
                    false, alo, false, blo, (short)0, accLo, false, false);
        accHi = __builtin_amdgcn_wmma_f32_16x16x4_f32(
                    false, ahi, false, bhi, (short)0, accHi, false, false);
    }

    float s = s2.x + s2.y;

    // Scalar tail (P not multiple of CHUNK) — handled once, by one lane.
    float tailDot = 0.0f;
    if (blockIdx.x == 0 && wave == 0 && lane == 0) {
        for (long long j = (long long)chunksPerSlice * CHUNK; j < (long long)P; ++j) {
            float p = pred[sliceBase + j], t = targ[sliceBase + j];
            tailDot += p * t;
            s += p + t;
        }
    }

    // Combine accumulators; diag(sum) == sum(diags).
    v8f acc = accLo + accHi;

    // Diagonal extraction: D VGPR v holds row v (lanes 0-15) and row v+8
    // (lanes 16-31). D[v][v] lives at lane v; D[v+8][v+8] at lane v+24.
    int sel = (lane < 8) ? lane : ((lane >= 24) ? (lane - 24) : -1);
    float r = acc[0];
    r = (sel == 1) ? acc[1] : r;
    r = (sel == 2) ? acc[2] : r;
    r = (sel == 3) ? acc[3] : r;
    r = (sel == 4) ? acc[4] : r;
    r = (sel == 5) ? acc[5] : r;
    r = (sel == 6) ? acc[6] : r;
    r = (sel == 7) ? acc[7] : r;
    float diag = (sel >= 0) ? r : 0.0f;
    diag += tailDot;

    // Wave32 butterfly reduce (fixed order -> deterministic).
    for (int m = 16; m >= 1; m >>= 1) {
        diag += __shfl_xor(diag, m, 32);
        s    += __shfl_xor(s,    m, 32);
    }

    __shared__ float sD[WAVES_PER_BLOCK];
    __shared__ float sS[WAVES_PER_BLOCK];
    if (lane == 0) { sD[wave] = diag; sS[wave] = s; }
    __syncthreads();
    if (threadIdx.x == 0) {
        float d = 0.0f, ss = 0.0f;
        for (int w = 0; w < WAVES_PER_BLOCK; ++w) { d += sD[w]; ss += sS[w]; }
        pDot[slice * gridDim.x + blockIdx.x] = d;
        pSum[slice * gridDim.x + blockIdx.x] = ss;
    }
}

__global__ __launch_bounds__(256) void dice_stage2(
    const float* __restrict__ pDot, const float* __restrict__ pSum,
    float* __restrict__ out)
{
    const int tid = threadIdx.x;
    const int nPart = NSLICES * BPS;      // 8192 partials per array
    __shared__ float red[256];
    __shared__ float inter[NSLICES];

    // Global sum S = sum(pred) + sum(target): fixed-order strided + tree.
    float a = 0.0f;
    for (int i = tid; i < nPart; i += 256) a += pSum[i];
    red[tid] = a;
    __syncthreads();
    for (int o = 128; o >= 1; o >>= 1) {
        if (tid < o) red[tid] += red[tid + o];
        __syncthreads();
    }
    float S = red[0];
    __syncthreads();

    // inter[bc]: 8 threads per slice, 32 partials each, fixed order.
    const int bc = tid >> 3, sub = tid & 7;
    float d = 0.0f;
    const int base = bc * BPS + sub * (BPS / 8);
    for (int i = 0; i < BPS / 8; ++i) d += pDot[base + i];
    red[tid] = d;
    __syncthreads();
    if (sub == 0) {
        float t = 0.0f;
        for (int k = 0; k < 8; ++k) t += red[bc * 8 + k];
        inter[bc] = t;
    }
    __syncthreads();

    if (tid == 0) {
        const float denom = S + 1.0f;                      // + SMOOTH
        // weights C^(b-N), C=8, N=4  -> 8^-4 .. 8^-1
        const float w[4] = {1.0f/4096.0f, 1.0f/512.0f, 1.0f/64.0f, 1.0f/8.0f};
        float total = 0.0f;
        for (int b = 0; b < 4; ++b) {
            float row = 0.0f;
            for (int c = 0; c < 8; ++c)
                row += 2.0f * (inter[b * 8 + c] + 1.0f) / denom;
            total += row * w[b];
        }
        total /= 4.0f;                                     // / N
        out[0] = 1.0f - total;
    }
}

extern "C" void kernel_launch(void* const* d_in, const int* in_sizes, int n_in,
                              void* d_out, int out_size, void* d_ws, size_t ws_size,
                              hipStream_t stream) {
    const float* pred = (const float*)d_in[0];
    const float* targ = (const float*)d_in[1];
    float* out = (float*)d_out;
    float* ws  = (float*)d_ws;

    const long long total = (long long)in_sizes[0];        // 4*8*2097152
    const int P      = (int)(total / NSLICES);
    const int chunks = P / CHUNK;

    float* pDot = ws;                        // NSLICES*BPS floats
    float* pSum = ws + NSLICES * BPS;        // NSLICES*BPS floats (64 KiB total)

    dice_stage1<<<dim3(BPS, NSLICES), 256, 0, stream>>>(pred, targ, pDot, pSum,
                                                        chunks, P);
    dice_stage2<<<1, 256, 0, stream>>>(pDot, pSum, out);
}